// OfficialLikeSpatialTransformer_6322191860354
// MI455X (gfx1250) — compile-verified
//
#include <hip/hip_runtime.h>

namespace {

constexpr int D = 160, H = 192, W = 224;
constexpr int HW = H * W;            // 43008
constexpr int N  = D * H * W;        // 6,881,280
constexpr int VPT = 2;               // voxels per thread along w (W even => no row wrap)
constexpr int BLOCK = 256;           // 8 wave32s per block
constexpr int NTHREADS = N / VPT;    // N even

// 8-byte vector, only 4-byte alignment guaranteed (gather base can be odd).
typedef float f32x2a4 __attribute__((ext_vector_type(2), aligned(4)));
// 8-byte vector, 8-byte aligned (flow/out pair accesses at even idx).
typedef float f32x2a8 __attribute__((ext_vector_type(2), aligned(8)));

__device__ __forceinline__ float lerp1(float a, float b, float t) {
  return __builtin_fmaf(t, b - a, a);
}

// One trilinear sample with the reference's zero-padding semantics.
__device__ __forceinline__ float sample_trilinear(const float* __restrict__ src,
                                                  float cd, float ch, float cw) {
  const float df = floorf(cd), hf = floorf(ch), wf = floorf(cw);
  const float fd = cd - df, fh = ch - hf, fw = cw - wf;
  const int i0 = (int)df, j0 = (int)hf, k0 = (int)wf;

  // Interior fast path: whole 2x2x2 stencil in bounds (vast majority since
  // flow ~ N(0,1)); unsigned compare also rejects negatives.
  const bool interior = ((unsigned)i0 <= (unsigned)(D - 2)) &&
                        ((unsigned)j0 <= (unsigned)(H - 2)) &&
                        ((unsigned)k0 <= (unsigned)(W - 2));
  if (interior) {
    const int base = (i0 * H + j0) * W + k0;
    // Two w-taps are adjacent: 4 paired b64 loads instead of 8 scalars.
    const f32x2a4 v00 = *(const f32x2a4*)(src + base);           // (d0,h0)
    const f32x2a4 v01 = *(const f32x2a4*)(src + base + W);       // (d0,h1)
    const f32x2a4 v10 = *(const f32x2a4*)(src + base + HW);      // (d1,h0)
    const f32x2a4 v11 = *(const f32x2a4*)(src + base + HW + W);  // (d1,h1)
    const float c00 = lerp1(v00.x, v00.y, fw);
    const float c01 = lerp1(v01.x, v01.y, fw);
    const float c10 = lerp1(v10.x, v10.y, fw);
    const float c11 = lerp1(v11.x, v11.y, fw);
    const float c0  = lerp1(c00, c01, fh);
    const float c1  = lerp1(c10, c11, fh);
    return lerp1(c0, c1, fd);
  }

  // Boundary path: 8 taps, clamped index, weight zeroed when out of bounds.
  float acc = 0.f;
#pragma unroll
  for (int dz = 0; dz < 2; ++dz) {
    const int id = i0 + dz;
    const float wd = dz ? fd : 1.f - fd;
    const bool bd = (unsigned)id < (unsigned)D;
    const int ic = min(max(id, 0), D - 1);
#pragma unroll
    for (int dy = 0; dy < 2; ++dy) {
      const int jh = j0 + dy;
      const float wh = dy ? fh : 1.f - fh;
      const bool bh = (unsigned)jh < (unsigned)H;
      const int jc = min(max(jh, 0), H - 1);
#pragma unroll
      for (int dx = 0; dx < 2; ++dx) {
        const int kw = k0 + dx;
        const float ww = dx ? fw : 1.f - fw;
        const bool bw = (unsigned)kw < (unsigned)W;
        const int kc = min(max(kw, 0), W - 1);
        const float wgt = (bd && bh && bw) ? (wd * wh * ww) : 0.f;
        acc = __builtin_fmaf(wgt, src[(ic * H + jc) * W + kc], acc);
      }
    }
  }
  return acc;
}

} // namespace

__global__ __launch_bounds__(BLOCK) void warp3d_trilinear_kernel(
    const float* __restrict__ src, const float* __restrict__ flow,
    float* __restrict__ out) {
  const int t = (int)(blockIdx.x * BLOCK + threadIdx.x);
  if (t >= NTHREADS) return;
  const int idx = t * VPT;  // even => 8B-aligned pair accesses, no row wrap

  // Prefetch the three streamed flow channels ~8 blocks ahead
  // (gfx1250 global_prefetch_b8; speculative, no counter cost).
  constexpr int PF = 8 * BLOCK * VPT;
  if (idx + PF < N) {
    __builtin_prefetch(&flow[idx + PF], 0, 0);
    __builtin_prefetch(&flow[idx + N + PF], 0, 0);
    __builtin_prefetch(&flow[idx + 2 * N + PF], 0, 0);
  }

  // idx -> (d, h, w) once per pair (compile-time constant div/mod).
  const unsigned u = (unsigned)idx;
  const int d = (int)(u / (unsigned)HW);
  const unsigned r = u % (unsigned)HW;
  const int h = (int)(r / (unsigned)W);
  const int w = (int)(r % (unsigned)W);  // even; w+1 < W always

  // Flow is streamed exactly once: paired nontemporal b64 loads so the
  // stream does not rinse src out of L2 (src = 27.5 MB resident in 192 MB L2).
  const f32x2a8 fD = __builtin_nontemporal_load((const f32x2a8*)(flow + idx));
  const f32x2a8 fH = __builtin_nontemporal_load((const f32x2a8*)(flow + idx + N));
  const f32x2a8 fW = __builtin_nontemporal_load((const f32x2a8*)(flow + idx + 2 * N));

  f32x2a8 res;
  res.x = sample_trilinear(src, (float)d + fD.x, (float)h + fH.x,
                           (float)w + fW.x);
  res.y = sample_trilinear(src, (float)d + fD.y, (float)h + fH.y,
                           (float)(w + 1) + fW.y);

  // Single nontemporal b64 store per pair; keep L2 for src.
  __builtin_nontemporal_store(res, (f32x2a8*)(out + idx));
}

extern "C" void kernel_launch(void* const* d_in, const int* in_sizes, int n_in,
                              void* d_out, int out_size, void* d_ws, size_t ws_size,
                              hipStream_t stream) {
  const float* src  = (const float*)d_in[0];   // (1,1,160,192,224) fp32
  const float* flow = (const float*)d_in[1];   // (1,3,160,192,224) fp32
  float* out = (float*)d_out;                  // (1,1,160,192,224) fp32
  const int blocks = (NTHREADS + BLOCK - 1) / BLOCK;  // 13440 blocks of 256
  warp3d_trilinear_kernel<<<blocks, BLOCK, 0, stream>>>(src, flow, out);
}